// LocalRelativePositionalEncoding_79061757985078
// MI455X (gfx1250) — compile-verified
//
#include <hip/hip_runtime.h>

// Problem constants (match reference)
#define BB   2
#define NP   2048
#define KN   16
#define NH   8
#define HIDN 64

typedef __attribute__((ext_vector_type(2))) float v2f;
typedef __attribute__((ext_vector_type(4))) float v4f;
typedef __attribute__((ext_vector_type(8))) float v8f;

// ---------------------------------------------------------------------------
// Kernel 1: stream 268 MB of zeros (the bandwidth-dominant phase).
// ---------------------------------------------------------------------------
__global__ __launch_bounds__(256) void zero_out_kernel(float* __restrict__ out, long n4) {
  long i = (long)blockIdx.x * blockDim.x + threadIdx.x;
  const long stride = (long)gridDim.x * blockDim.x;
  v4f z = {0.0f, 0.0f, 0.0f, 0.0f};
  v4f* p = (v4f*)out;
  for (; i < n4; i += stride) {
    __builtin_nontemporal_store(z, p + i);   // global_store_b128, NT hint
  }
}

// ---------------------------------------------------------------------------
// Kernel 2: exact top-K=16 nearest neighbors, one wave32 per query point.
// Per lane: scan 64 candidates, keep sorted top-16 in registers (static
// indices only -> VGPR resident). Then 32-way merge of sorted lists via LDS
// + 16 rounds of shfl_xor min-reduction.
// ---------------------------------------------------------------------------
__global__ __launch_bounds__(256) void topk_kernel(const float* __restrict__ xyz,
                                                   int* __restrict__ sel) {
  const int lane = threadIdx.x & 31;
  const int wave = threadIdx.x >> 5;
  const int q = blockIdx.x * 8 + wave;          // query id in [0, B*N)
  const int b = q / NP, i = q % NP;
  const float* __restrict__ X = xyz + (size_t)b * 3 * NP;
  const float xi = X[i], yi = X[NP + i], zi = X[2 * NP + i];

  float dist[KN];
  int   ind[KN];
#pragma unroll
  for (int s = 0; s < KN; ++s) { dist[s] = 3.0e38f; ind[s] = 0; }

  for (int j = lane; j < NP; j += 32) {
    const float dx = xi - X[j];
    const float dy = yi - X[NP + j];
    const float dz = zi - X[2 * NP + j];
    const float d2 = dx * dx + dy * dy + dz * dz;
    if (d2 < dist[KN - 1]) {
      dist[KN - 1] = d2; ind[KN - 1] = j;
#pragma unroll
      for (int s = KN - 1; s > 0; --s) {        // bubble-up, all static indices
        if (dist[s] < dist[s - 1]) {
          float td = dist[s]; dist[s] = dist[s - 1]; dist[s - 1] = td;
          int   ti = ind[s];  ind[s]  = ind[s - 1];  ind[s - 1]  = ti;
        }
      }
    }
  }

  __shared__ float sd[8][32 * KN];   // 16 KB
  __shared__ int   si[8][32 * KN];   // 16 KB
#pragma unroll
  for (int s = 0; s < KN; ++s) {
    sd[wave][lane * KN + s] = dist[s];
    si[wave][lane * KN + s] = ind[s];
  }
  __syncthreads();

  int head = 0, selv = 0;
  for (int r = 0; r < KN; ++r) {
    float cd = (head < KN) ? sd[wave][lane * KN + head] : 3.0e38f;
    int   ci = (head < KN) ? si[wave][lane * KN + head] : 0;
    int   cl = lane;
#pragma unroll
    for (int off = 16; off >= 1; off >>= 1) {   // wave32 min-reduce (d2, lane, idx)
      float od = __shfl_xor(cd, off, 32);
      int   oi = __shfl_xor(ci, off, 32);
      int   ol = __shfl_xor(cl, off, 32);
      if (od < cd || (od == cd && ol < cl)) { cd = od; ci = oi; cl = ol; }
    }
    if (lane == r) selv = ci;                   // lane r keeps the r-th winner
    if (lane == cl) head++;                     // winner's list advances
  }
  if (lane < KN) sel[(size_t)q * KN + lane] = selv;
}

// ---------------------------------------------------------------------------
// Kernel 3: MLP (3 -> 64 -> 8) on the 16 neighbors of one query via
// v_wmma_f32_16x16x4_f32, then scatter 8 floats per (query, neighbor).
// One wave32 per query. 20 WMMAs per wave. All fragment loads are
// branch-free (address-select / index-clamp + cndmask) so no exec-mask
// save/restore interleaves with the WMMA chain.
// A 16x4 f32 layout: lane l holds row M=l%16, K = (l<16 ? {0,1} : {2,3}).
// C/D 16x16 f32: reg r on lane l = (M = r + 8*(l>=16), N = l%16).
// ---------------------------------------------------------------------------
__global__ __launch_bounds__(32) void mlp_scatter_kernel(
    const float* __restrict__ xyz, const int* __restrict__ sel,
    const float* __restrict__ w1, const float* __restrict__ b1,
    const float* __restrict__ w2, const float* __restrict__ b2,
    float* __restrict__ out) {
  const int lane = threadIdx.x;     // 0..31, full wave (EXEC all ones for WMMA)
  const int m  = lane & 15;
  const int hi = lane >> 4;         // 0: K={0,1} half, 1: K={2,3} half
  const int q = blockIdx.x;
  const int b = q / NP, i = q % NP;
  const float* __restrict__ X = xyz + (size_t)b * 3 * NP;

  // rel_pos for neighbor row m (both wave halves need row M = l%16)
  const int jm = sel[(size_t)q * KN + m];
  const float relx = X[i]          - X[jm];
  const float rely = X[NP + i]     - X[NP + jm];
  const float relz = X[2 * NP + i] - X[2 * NP + jm];

  // A fragment: rel_pos padded [16 x 4] (K=3 column is zero)
  v2f a;
  a.x = hi ? relz : relx;           // values already in regs -> cndmask only
  a.y = hi ? 0.0f : rely;

  __shared__ float hbuf[16 * HIDN];    // 4 KB: h matrix, row-major [16][64]

  const int w1row0 = hi ? 2 : 0;       // address select, single unconditional load
  const int mc     = (m < NH) ? m : 0; // clamped column for in-bounds w2/b2 loads
  const bool mval  = (m < NH);

  // ---- layer 1: h = relu(rel @ w1 + b1), 4 WMMA tiles over N=64 ----
#pragma unroll
  for (int t = 0; t < 4; ++t) {
    const int n = t * 16 + m;
    v2f bf;                                       // w1 padded [4 x 16] tile
    bf.x = w1[w1row0 * HIDN + n];                 // row 0 or row 2
    const float w1r1 = w1[1 * HIDN + n];          // unconditional load
    bf.y = hi ? 0.0f : w1r1;                      // K=3 pad -> cndmask
    const float bias1 = b1[n];                    // bias depends on N only
    v8f c;
#pragma unroll
    for (int r = 0; r < 8; ++r) c[r] = bias1;
    c = __builtin_amdgcn_wmma_f32_16x16x4_f32(false, a, false, bf,
                                              (short)0, c, false, false);
#pragma unroll
    for (int r = 0; r < 8; ++r) {                 // single v_max_num_f32 ReLU
      hbuf[(r + 8 * hi) * HIDN + n] = fmaxf(c[r], 0.0f);
    }
  }
  __syncthreads();   // single-wave WG: lowers to DS-count wait, orders LDS RAW

  // ---- layer 2: bias = h @ w2pad + b2pad, 16 chained WMMAs over K=64 ----
  v8f acc;
  const float bias2 = mval ? b2[mc] : 0.0f;       // clamped load + cndmask
#pragma unroll
  for (int r = 0; r < 8; ++r) acc[r] = bias2;

#pragma unroll
  for (int s = 0; s < 16; ++s) {
    const int k0 = 4 * s + 2 * hi;                // this lane's two K rows
    const v2f a2 = *(const v2f*)(hbuf + m * HIDN + k0);   // 8B-aligned ds_load_b64
    const float w2a = w2[(k0 + 0) * NH + mc];     // in-bounds, unconditional
    const float w2b = w2[(k0 + 1) * NH + mc];
    v2f b2f;                                      // w2 padded [64 x 16], cols >=8 zero
    b2f.x = mval ? w2a : 0.0f;
    b2f.y = mval ? w2b : 0.0f;
    acc = __builtin_amdgcn_wmma_f32_16x16x4_f32(false, a2, false, b2f,
                                                (short)0, acc, false, false);
  }

  // ---- scatter into pre-zeroed out[B,H,N,N] ----
  if (mval) {                                     // N = head index, valid < 8
#pragma unroll
    for (int r = 0; r < 8; ++r) {
      const int row = r + 8 * hi;                 // neighbor slot M
      const int jn  = sel[(size_t)q * KN + row];
      out[(((size_t)b * NH + m) * NP + i) * NP + jn] = acc[r];
    }
  }
}

// ---------------------------------------------------------------------------
extern "C" void kernel_launch(void* const* d_in, const int* in_sizes, int n_in,
                              void* d_out, int out_size, void* d_ws, size_t ws_size,
                              hipStream_t stream) {
  (void)in_sizes; (void)n_in; (void)ws_size;
  const float* xyz = (const float*)d_in[0];   // [B,3,N]
  const float* w1  = (const float*)d_in[1];   // [3,64]
  const float* b1  = (const float*)d_in[2];   // [64]
  const float* w2  = (const float*)d_in[3];   // [64,8]
  const float* b2  = (const float*)d_in[4];   // [8]
  float* out = (float*)d_out;                 // [B,H,N,N]
  int*   sel = (int*)d_ws;                    // [B,N,K] neighbor indices (256 KB)

  const long n4 = (long)out_size / 4;         // 16,777,216 float4 stores
  zero_out_kernel<<<4096, 256, 0, stream>>>(out, n4);
  topk_kernel<<<(BB * NP) / 8, 256, 0, stream>>>(xyz, sel);
  mlp_scatter_kernel<<<BB * NP, 32, 0, stream>>>(xyz, sel, w1, b1, w2, b2, out);
}